// ProductQuantization_41764261987121
// MI455X (gfx1250) — compile-verified
//
#include <hip/hip_runtime.h>

typedef __attribute__((ext_vector_type(2))) float v2f;
typedef __attribute__((ext_vector_type(8))) float v8f;

#define PQ_K 8
#define PQ_C 256
#define PQ_SD 16
#define PQ_D 128

// ---------------------------------------------------------------------------
// Kernel 1: precompute c_sq[k][c] = sum_d codebook[k][c][d]^2  (2048 floats)
// ---------------------------------------------------------------------------
__global__ __launch_bounds__(256) void pq_csq_kernel(const float* __restrict__ cb,
                                                     float* __restrict__ csq) {
    int id = blockIdx.x * blockDim.x + threadIdx.x;   // 0 .. K*C-1
    if (id >= PQ_K * PQ_C) return;
    const float* row = cb + (size_t)id * PQ_SD;
    float s = 0.f;
#pragma unroll
    for (int d = 0; d < PQ_SD; ++d) s += row[d] * row[d];
    csq[id] = s;
}

// ---------------------------------------------------------------------------
// Kernel 2: per block: 16-vector tile; wave w handles subquantizer k = w.
// Cross term via V_WMMA_F32_16X16X4_F32 (exact fp32); argmin via a
// branchless u64 (ordered-dist | index) min butterfly over 16-lane halves.
// ---------------------------------------------------------------------------
__global__ __launch_bounds__(256) void pq_main_kernel(const float* __restrict__ vec,
                                                      const float* __restrict__ cb,
                                                      const float* __restrict__ csq,
                                                      int* __restrict__ out_idx,
                                                      float* __restrict__ out_rec) {
    const int lane  = threadIdx.x & 31;
    const int k     = threadIdx.x >> 5;          // wave id == subquantizer
    const int n0    = blockIdx.x * 16;           // 16 vectors per block
    const int m     = lane & 15;                 // row/col within 16
    const int kbase = (lane >> 4) * 2;           // K-pair offset for A/B frags

    // --- A fragments: vectors [n0..n0+15], dims [k*16 .. k*16+15], 4 chunks of K=4.
    // 32-bit A 16x4 layout: lanes 0-15 hold M=lane, K={0,1}; lanes 16-31 K={2,3}.
    const float* vrow = vec + (size_t)(n0 + m) * PQ_D + k * PQ_SD;
    v2f a[4];
#pragma unroll
    for (int kk = 0; kk < 4; ++kk)
        a[kk] = *(const v2f*)(vrow + kk * 4 + kbase);

    const float* cbk  = cb  + (size_t)k * PQ_C * PQ_SD;
    const float* csqk = csq + k * PQ_C;

    float best[8];
    int   bidx[8];
#pragma unroll
    for (int r = 0; r < 8; ++r) { best[r] = 3.4e38f; bidx[r] = 0; }

    // --- sweep 16 centroid tiles of 16
    for (int c0 = 0; c0 < PQ_C; c0 += 16) {
        v8f acc = {};
        // B fragment: B[K][N] = codebook[c0+N][K]; lane L serves column L%16,
        // K-pair (L/16)*2 -> consecutive floats in the codebook row.
        const float* brow = cbk + (size_t)(c0 + m) * PQ_SD + kbase;
#pragma unroll
        for (int kk = 0; kk < 4; ++kk) {
            v2f b = *(const v2f*)(brow + kk * 4);
            acc = __builtin_amdgcn_wmma_f32_16x16x4_f32(
                /*neg_a=*/false, a[kk], /*neg_b=*/false, b,
                /*c_mod=*/(short)0, acc, /*reuse_a=*/false, /*reuse_b=*/false);
        }
        // C/D layout: VGPR r, lanes 0-15 -> M=r, N=lane; lanes 16-31 -> M=r+8.
        // This lane's column is c = c0 + m for every r.
        const float cs = csqk[c0 + m];
        const int   c  = c0 + m;
#pragma unroll
        for (int r = 0; r < 8; ++r) {
            float d = cs - 2.0f * acc[r];    // v^2 term irrelevant for argmin
            if (d < best[r]) { best[r] = d; bidx[r] = c; }   // strict < : first min
        }
    }

    // --- pack (ordered-dist, idx) into u64; branchless min butterfly over
    //     the 16 lanes of each half. Low word = index => ties pick smaller c.
    unsigned long long key[8];
#pragma unroll
    for (int r = 0; r < 8; ++r) {
        unsigned u = __float_as_uint(best[r]);
        u = (u & 0x80000000u) ? ~u : (u | 0x80000000u);   // monotonic map
        key[r] = ((unsigned long long)u << 32) | (unsigned)bidx[r];
    }
#pragma unroll
    for (int s = 1; s < 16; s <<= 1) {
#pragma unroll
        for (int r = 0; r < 8; ++r) {
            unsigned long long ok =
                (unsigned long long)__shfl_xor((long long)key[r], s, 32);
            key[r] = (ok < key[r]) ? ok : key[r];
        }
    }

    // --- emit: per register r, lanes 0-15 hold result for M=r, lanes 16-31
    //     for M=r+8. Each 16-lane half writes its own row: 8 iterations.
    const int half = lane >> 4;
    const int col  = lane & 15;
    for (int mm = 0; mm < 8; ++mm) {
        int idx0 = (int)(unsigned)__shfl((long long)key[mm], 0, 32);   // M = mm
        int idx1 = (int)(unsigned)__shfl((long long)key[mm], 16, 32);  // M = mm+8
        int idx  = half ? idx1 : idx0;
        int row  = mm + half * 8;
        if (col == 0)
            out_idx[(size_t)(n0 + row) * PQ_K + k] = idx;
        out_rec[(size_t)(n0 + row) * PQ_D + k * PQ_SD + col] =
            cbk[(size_t)idx * PQ_SD + col];
    }
}

// ---------------------------------------------------------------------------
extern "C" void kernel_launch(void* const* d_in, const int* in_sizes, int n_in,
                              void* d_out, int out_size, void* d_ws, size_t ws_size,
                              hipStream_t stream) {
    const float* vec = (const float*)d_in[0];   // [N, 128] f32
    const float* cb  = (const float*)d_in[1];   // [8, 256, 16] f32
    const int N = in_sizes[0] / PQ_D;

    float* csq = (float*)d_ws;                  // 2048 floats scratch

    int* out_idx   = (int*)d_out;               // first N*K int32 (bitwise)
    float* out_rec = (float*)d_out + (size_t)N * PQ_K;

    pq_csq_kernel<<<(PQ_K * PQ_C + 255) / 256, 256, 0, stream>>>(cb, csq);
    pq_main_kernel<<<N / 16, 256, 0, stream>>>(vec, cb, csq, out_idx, out_rec);
}